// EdgeUpdateEncoder_60120952209606
// MI455X (gfx1250) — compile-verified
//
#include <hip/hip_runtime.h>

// Problem constants (from the reference)
#define NN 50000   // nodes
#define EE 200000  // edges (divisible by 16)
#define BB 64      // graphs
#define LL 6       // message-passing layers

typedef float v2f __attribute__((ext_vector_type(2)));
typedef float v8f __attribute__((ext_vector_type(8)));

__device__ inline v2f ld2(const float* p) { return *(const v2f*)p; }

// ---------------------------------------------------------------------------
// Wave-level 16x16 f32 GEMM tile on V_WMMA_F32_16X16X4_F32.
//   A (16x4 f32): lanes 0-15 hold M=0..15 K{0,1}; lanes 16-31 K{2,3}.
//   B (4x16): VGPR0 = row K0 (lanes 0-15) / K2 (lanes 16-31); VGPR1 = K1/K3.
//   C/D: VGPR r -> M=r (lanes 0-15) / M=r+8 (lanes 16-31); N = lane&15.
// fa(row, kk, h2) returns A[row][kk+h2 .. kk+h2+1] as v2f (8B-aligned load),
// with kk a compile-time constant (segment branches fold) and h2 in {0,2}.
// ---------------------------------------------------------------------------
template <int K, typename FA>
__device__ inline v8f tile_gemm(FA fa, const float* __restrict__ B, int ldb,
                                int col0, v8f acc) {
  const int lane = threadIdx.x & 31;
  const int h2   = (lane >> 4) * 2;
  const int l15  = lane & 15;
#pragma unroll
  for (int kk = 0; kk < K; kk += 4) {
    v2f a = fa(l15, kk, h2);
    v2f b;
    const float* bp = B + (size_t)(kk + h2) * ldb + col0 + l15;
    b.x = bp[0];
    b.y = bp[ldb];
    acc = __builtin_amdgcn_wmma_f32_16x16x4_f32(false, a, false, b,
                                                (short)0, acc, false, false);
  }
  return acc;
}

template <typename ST>
__device__ inline void tile_store(v8f acc, ST st) {
  const int lane = threadIdx.x & 31;
  const int half = lane >> 4;
  const int l15  = lane & 15;
#pragma unroll
  for (int r = 0; r < 8; ++r) st(r + half * 8, l15, acc[r]);
}

// ---------------------------------------------------------------------------
// Encoders: out[M,NC] = A[M,K] @ W[K,NC] + bias. A tile staged once in LDS
// so the NC/16 waves of the block share it instead of re-reading global.
// ---------------------------------------------------------------------------
template <int K, int NC>
__global__ void encoder_kernel(const float* __restrict__ A,
                               const float* __restrict__ W,
                               const float* __restrict__ bias,
                               float* __restrict__ out) {
  __shared__ __align__(16) float a_s[16 * K];
  const int m0 = blockIdx.x * 16;
  for (int i = threadIdx.x; i < 16 * K / 4; i += blockDim.x) {
    int r = i / (K / 4), c = (i % (K / 4)) * 4;
    *(float4*)(a_s + r * K + c) = *(const float4*)(A + (size_t)(m0 + r) * K + c);
  }
  __syncthreads();
  const int col0 = (threadIdx.x >> 5) * 16;
  v8f acc = {};
  acc = tile_gemm<K>([&](int r, int kk, int h2) {
    return ld2(a_s + r * K + kk + h2);
  }, W, NC, col0, acc);
  tile_store(acc, [&](int r, int c, float v) {
    out[(size_t)(m0 + r) * NC + col0 + c] = v + bias[col0 + c];
  });
}

// ---------------------------------------------------------------------------
// Fused edge kernel (one block = 16 edges, 4 waves):
//   stage G:  gather hs=h[src], hd=h[dst], e tile -> LDS (once per edge)
//   stage 0:  t = e + hs @ mw                     (K=128, from LDS)
//   stage 1:  u = relu([hs|hd|t] @ w1 + b1)       (K=320, from LDS)
//   stage 2:  e_new = e + u @ w2 + b2; store e; atomic scatter into macc[dst]
// ---------------------------------------------------------------------------
__global__ void edge_mlp_kernel(const float* __restrict__ h, float* __restrict__ e,
                                float* __restrict__ macc,
                                const int* __restrict__ src, const int* __restrict__ dst,
                                const float* __restrict__ mw,
                                const float* __restrict__ w1, const float* __restrict__ b1,
                                const float* __restrict__ w2, const float* __restrict__ b2) {
  __shared__ __align__(16) float hs_s[16 * 128];
  __shared__ __align__(16) float hd_s[16 * 128];
  __shared__ __align__(16) float e_s[16 * 64];
  __shared__ __align__(16) float t_s[16 * 64];
  __shared__ __align__(16) float u_s[16 * 64];
  const int m0   = blockIdx.x * 16;
  const int col0 = (threadIdx.x >> 5) * 16;  // 4 waves -> 64 cols

  for (int i = threadIdx.x; i < 16 * 128 / 4; i += blockDim.x) {
    int r = i >> 5, c = (i & 31) * 4;
    *(float4*)(hs_s + r * 128 + c) = *(const float4*)(h + (size_t)src[m0 + r] * 128 + c);
    *(float4*)(hd_s + r * 128 + c) = *(const float4*)(h + (size_t)dst[m0 + r] * 128 + c);
  }
  for (int i = threadIdx.x; i < 16 * 64 / 4; i += blockDim.x) {
    int r = i >> 4, c = (i & 15) * 4;
    *(float4*)(e_s + r * 64 + c) = *(const float4*)(e + (size_t)(m0 + r) * 64 + c);
  }
  __syncthreads();

  // stage 0: message  t = e + hs @ mw
  v8f acc0 = {};
  acc0 = tile_gemm<128>([&](int r, int kk, int h2) {
    return ld2(hs_s + r * 128 + kk + h2);
  }, mw, 64, col0, acc0);
  tile_store(acc0, [&](int r, int c, float v) {
    t_s[r * 64 + col0 + c] = e_s[r * 64 + col0 + c] + v;
  });
  __syncthreads();

  // stage 1: hidden  u = relu([hs|hd|t] @ w1 + b1), K = 320
  v8f acc1 = {};
  acc1 = tile_gemm<320>([&](int r, int kk, int h2) {
    if (kk < 128) return ld2(hs_s + r * 128 + kk + h2);
    if (kk < 256) return ld2(hd_s + r * 128 + (kk - 128) + h2);
    return ld2(t_s + r * 64 + (kk - 256) + h2);
  }, w1, 64, col0, acc1);
  tile_store(acc1, [&](int r, int c, float v) {
    float t = v + b1[col0 + c];
    u_s[r * 64 + col0 + c] = t > 0.f ? t : 0.f;
  });
  __syncthreads();

  // stage 2: e_new = e + u @ w2 + b2; scatter-add into macc[dst]
  v8f acc2 = {};
  acc2 = tile_gemm<64>([&](int r, int kk, int h2) {
    return ld2(u_s + r * 64 + kk + h2);
  }, w2, 64, col0, acc2);
  tile_store(acc2, [&](int r, int c, float v) {
    int erow = m0 + r;
    float val = e_s[r * 64 + col0 + c] + v + b2[col0 + c];
    e[(size_t)erow * 64 + col0 + c] = val;
    atomicAdd(macc + (size_t)dst[erow] * 64 + col0 + c, val);
  });
}

// ---------------------------------------------------------------------------
// Fused node kernel (one block = 16 nodes, 8 waves):
//   stage G: h tile + (macc*rdeg) tile -> LDS
//   stage 1: u = relu([h|m] @ w1 + b1)  (K=192, waves 0-3)
//   stage 2: h += u @ w2 + b2           (K=64, 8 waves -> 128 cols)
// ---------------------------------------------------------------------------
__global__ void node_mlp_kernel(float* __restrict__ h, const float* __restrict__ macc,
                                const float* __restrict__ rdeg,
                                const float* __restrict__ w1, const float* __restrict__ b1,
                                const float* __restrict__ w2, const float* __restrict__ b2) {
  __shared__ __align__(16) float hv_s[16 * 128];
  __shared__ __align__(16) float m_s[16 * 64];
  __shared__ __align__(16) float u_s[16 * 64];
  const int m0   = blockIdx.x * 16;
  const int wave = threadIdx.x >> 5;   // 8 waves

  for (int i = threadIdx.x; i < 16 * 128 / 4; i += blockDim.x) {
    int r = i >> 5, c = (i & 31) * 4;
    *(float4*)(hv_s + r * 128 + c) = *(const float4*)(h + (size_t)(m0 + r) * 128 + c);
  }
  for (int i = threadIdx.x; i < 16 * 64 / 4; i += blockDim.x) {
    int r = i >> 4, c = (i & 15) * 4;
    float4 v = *(const float4*)(macc + (size_t)(m0 + r) * 64 + c);
    float rd = rdeg[m0 + r];                     // scatter_mean divide
    v.x *= rd; v.y *= rd; v.z *= rd; v.w *= rd;
    *(float4*)(m_s + r * 64 + c) = v;
  }
  __syncthreads();

  if (wave < 4) {                      // wave-uniform: EXEC stays all-ones
    const int col0 = wave * 16;
    v8f acc = {};
    acc = tile_gemm<192>([&](int r, int kk, int h2) {
      if (kk < 128) return ld2(hv_s + r * 128 + kk + h2);
      return ld2(m_s + r * 64 + (kk - 128) + h2);
    }, w1, 64, col0, acc);
    tile_store(acc, [&](int r, int c, float v) {
      float t = v + b1[col0 + c];
      u_s[r * 64 + col0 + c] = t > 0.f ? t : 0.f;
    });
  }
  __syncthreads();

  const int col0 = wave * 16;          // 8 waves -> 128 output cols
  v8f acc2 = {};
  acc2 = tile_gemm<64>([&](int r, int kk, int h2) {
    return ld2(u_s + r * 64 + kk + h2);
  }, w2, 128, col0, acc2);
  tile_store(acc2, [&](int r, int c, float v) {
    h[(size_t)(m0 + r) * 128 + col0 + c] = hv_s[r * 128 + col0 + c] + v + b2[col0 + c];
  });
}

// --------------------------- small helper kernels ---------------------------
__global__ void deg_count_kernel(const int* __restrict__ dst, float* __restrict__ cnt) {
  int i = blockIdx.x * blockDim.x + threadIdx.x;
  if (i < EE) atomicAdd(cnt + dst[i], 1.0f);
}

__global__ void recip_clamp_kernel(float* __restrict__ v, int n) {
  int i = blockIdx.x * blockDim.x + threadIdx.x;
  if (i < n) v[i] = 1.0f / fmaxf(v[i], 1.0f);
}

__global__ void pool_kernel(const float* __restrict__ h, const int* __restrict__ batch,
                            float* __restrict__ gsum, float* __restrict__ gcnt) {
  int i = blockIdx.x * blockDim.x + threadIdx.x;   // NN*128 threads
  int node = i >> 7, c = i & 127;
  if (node < NN) {
    int g = batch[node];
    atomicAdd(gsum + (size_t)g * 128 + c, h[(size_t)node * 128 + c]);
    if (c == 0) atomicAdd(gcnt + g, 1.0f);
  }
}

// out = relu((gsum*rgcnt) @ w1 + b1) @ w2 + b2   (64x128, fused via LDS)
__global__ void readout_kernel(const float* __restrict__ gsum, const float* __restrict__ rgcnt,
                               const float* __restrict__ w1, const float* __restrict__ b1,
                               const float* __restrict__ w2, const float* __restrict__ b2,
                               float* __restrict__ out) {
  __shared__ __align__(16) float g_s[16 * 128];
  __shared__ __align__(16) float t_s[16 * 128];
  const int m0   = blockIdx.x * 16;
  const int col0 = (threadIdx.x >> 5) * 16;   // 8 waves -> 128 cols

  for (int i = threadIdx.x; i < 16 * 128 / 4; i += blockDim.x) {
    int r = i >> 5, c = (i & 31) * 4;
    float4 v = *(const float4*)(gsum + (size_t)(m0 + r) * 128 + c);
    float rc = rgcnt[m0 + r];
    v.x *= rc; v.y *= rc; v.z *= rc; v.w *= rc;
    *(float4*)(g_s + r * 128 + c) = v;
  }
  __syncthreads();

  v8f acc = {};
  acc = tile_gemm<128>([&](int r, int kk, int h2) {
    return ld2(g_s + r * 128 + kk + h2);
  }, w1, 128, col0, acc);
  tile_store(acc, [&](int r, int c, float v) {
    float t = v + b1[col0 + c];
    t_s[r * 128 + col0 + c] = t > 0.f ? t : 0.f;
  });
  __syncthreads();

  v8f acc2 = {};
  acc2 = tile_gemm<128>([&](int r, int kk, int h2) {
    return ld2(t_s + r * 128 + kk + h2);
  }, w2, 128, col0, acc2);
  tile_store(acc2, [&](int r, int c, float v) {
    out[(size_t)(m0 + r) * 128 + col0 + c] = v + b2[col0 + c];
  });
}

// ---------------------------------------------------------------------------
extern "C" void kernel_launch(void* const* d_in, const int* in_sizes, int n_in,
                              void* d_out, int out_size, void* d_ws, size_t ws_size,
                              hipStream_t stream) {
  (void)in_sizes; (void)n_in; (void)out_size; (void)ws_size;
  const float* x       = (const float*)d_in[0];
  const float* ea      = (const float*)d_in[1];
  const int*   eidx    = (const int*)d_in[2];
  const int*   batch   = (const int*)d_in[3];
  const float* lin_x_w = (const float*)d_in[4];
  const float* lin_x_b = (const float*)d_in[5];
  const float* lin_e_w = (const float*)d_in[6];
  const float* lin_e_b = (const float*)d_in[7];
  const float* msg_w   = (const float*)d_in[8];
  const float* phie_w1 = (const float*)d_in[9];
  const float* phie_b1 = (const float*)d_in[10];
  const float* phie_w2 = (const float*)d_in[11];
  const float* phie_b2 = (const float*)d_in[12];
  const float* phiv_w1 = (const float*)d_in[13];
  const float* phiv_b1 = (const float*)d_in[14];
  const float* phiv_w2 = (const float*)d_in[15];
  const float* phiv_b2 = (const float*)d_in[16];
  const float* ro_w1   = (const float*)d_in[17];
  const float* ro_b1   = (const float*)d_in[18];
  const float* ro_w2   = (const float*)d_in[19];
  const float* ro_b2   = (const float*)d_in[20];
  const int* src = eidx;        // edge_index[0]
  const int* dst = eidx + EE;   // edge_index[1]

  // Workspace carve-out (~90 MB): h, e, macc, rdeg, gsum, gcnt
  char* wp = (char*)d_ws;
  auto carve = [&](size_t bytes) {
    char* p = wp; wp += (bytes + 255) & ~(size_t)255; return p;
  };
  float* h    = (float*)carve((size_t)NN * 128 * 4);
  float* e    = (float*)carve((size_t)EE * 64 * 4);
  float* macc = (float*)carve((size_t)NN * 64 * 4);
  float* rdeg = (float*)carve((size_t)NN * 4);
  float* gsum = (float*)carve((size_t)BB * 128 * 4);
  float* gcnt = (float*)carve((size_t)BB * 4);

  // Encoders
  encoder_kernel<64, 128><<<NN / 16, 256, 0, stream>>>(x, lin_x_w, lin_x_b, h);
  encoder_kernel<16, 64><<<EE / 16, 128, 0, stream>>>(ea, lin_e_w, lin_e_b, e);

  // Reciprocal clamped in-degree
  hipMemsetAsync(rdeg, 0, (size_t)NN * 4, stream);
  deg_count_kernel<<<(EE + 255) / 256, 256, 0, stream>>>(dst, rdeg);
  recip_clamp_kernel<<<(NN + 255) / 256, 256, 0, stream>>>(rdeg, NN);

  // Message-passing layers (msg GEMM fused into edge kernel stage 0)
  for (int l = 0; l < LL; ++l) {
    hipMemsetAsync(macc, 0, (size_t)NN * 64 * 4, stream);
    edge_mlp_kernel<<<EE / 16, 128, 0, stream>>>(h, e, macc, src, dst,
        msg_w + (size_t)l * 128 * 64,
        phie_w1 + (size_t)l * 320 * 64, phie_b1 + (size_t)l * 64,
        phie_w2 + (size_t)l * 64 * 64,  phie_b2 + (size_t)l * 64);
    node_mlp_kernel<<<NN / 16, 256, 0, stream>>>(h, macc, rdeg,
        phiv_w1 + (size_t)l * 192 * 64, phiv_b1 + (size_t)l * 64,
        phiv_w2 + (size_t)l * 64 * 128, phiv_b2 + (size_t)l * 128);
  }

  // Graph readout
  hipMemsetAsync(gsum, 0, (size_t)BB * 128 * 4, stream);
  hipMemsetAsync(gcnt, 0, (size_t)BB * 4, stream);
  pool_kernel<<<(NN * 128 + 255) / 256, 256, 0, stream>>>(h, batch, gsum, gcnt);
  recip_clamp_kernel<<<1, BB, 0, stream>>>(gcnt, BB);
  readout_kernel<<<BB / 16, 256, 0, stream>>>(gsum, gcnt, ro_w1, ro_b1, ro_w2, ro_b2,
                                              (float*)d_out);
}